// MyAGNN_new_60241211293939
// MI455X (gfx1250) — compile-verified
//
#include <hip/hip_runtime.h>

// ---------------------------------------------------------------------------
// AGNN forward for MI455X (gfx1250, wave32).
// Dense GEMMs use V_WMMA_F32_16X16X4_F32 (fp32-exact matrix path).
// Graph phases are L2-resident gather/scatter (h = 20.5MB << 192MB L2).
// ---------------------------------------------------------------------------

typedef float v2f __attribute__((ext_vector_type(2)));
typedef float v8f __attribute__((ext_vector_type(8)));

#define N_NODES 20000
#define N_EDGES 640000
#define IN_DIM  128
#define HIDDEN  256
#define OUT_DIM 64

// ---------------------------------------------------------------------------
// GEMM: C[M,N] = act(A[M,K] @ W[K,N] + bias), row-major, fp32 WMMA.
// One wave computes a 16x64 tile: 4 N-tiles sharing one A fragment per K-step.
// Requires M%16==0, N%64==0, K%4==0 (true for all call sites).
// ---------------------------------------------------------------------------
__global__ void gemm_wmma_f32_kernel(const float* __restrict__ A,
                                     const float* __restrict__ W,
                                     const float* __restrict__ bias,
                                     float* __restrict__ C,
                                     int M, int K, int N, int do_relu)
{
    const int lane   = threadIdx.x & 31;
    const int waveId = (blockIdx.x * blockDim.x + threadIdx.x) >> 5;
    const int nGroups = N >> 6;                 // groups of 64 columns
    const int mTile  = waveId / nGroups;
    const int nGroup = waveId % nGroups;
    if (mTile * 16 >= M) return;                // wave-uniform exit (EXEC stays full)

    const int m0  = mTile * 16;
    const int n0  = nGroup * 64;
    const int r16 = lane & 15;                  // row-in-tile (A) / col-in-tile (B,C)
    const int kb  = (lane >> 4) << 1;           // k-pair base within K-step of 4

    v8f acc0 = {0.f,0.f,0.f,0.f,0.f,0.f,0.f,0.f};
    v8f acc1 = acc0, acc2 = acc0, acc3 = acc0;

    const float* arow = A + (size_t)(m0 + r16) * K + kb;
    for (int k0 = 0; k0 < K; k0 += 4) {
        v2f a;
        a.x = arow[k0];
        a.y = arow[k0 + 1];

        const float* wp = W + (size_t)(k0 + kb) * N + n0 + r16;
        v2f b0, b1, b2, b3;
        b0.x = wp[0];      b0.y = wp[N];
        b1.x = wp[16];     b1.y = wp[N + 16];
        b2.x = wp[32];     b2.y = wp[N + 32];
        b3.x = wp[48];     b3.y = wp[N + 48];

        acc0 = __builtin_amdgcn_wmma_f32_16x16x4_f32(false, a, false, b0, (short)0, acc0, false, false);
        acc1 = __builtin_amdgcn_wmma_f32_16x16x4_f32(false, a, false, b1, (short)0, acc1, false, false);
        acc2 = __builtin_amdgcn_wmma_f32_16x16x4_f32(false, a, false, b2, (short)0, acc2, false, false);
        acc3 = __builtin_amdgcn_wmma_f32_16x16x4_f32(false, a, false, b3, (short)0, acc3, false, false);
    }

    // C/D layout: VGPR r, lane l -> row m0 + r + 8*(l>>4), col n0 + t*16 + (l&15)
    const int mBase = m0 + ((lane >> 4) << 3);
    v8f accs[4] = {acc0, acc1, acc2, acc3};
    #pragma unroll
    for (int t = 0; t < 4; ++t) {
        const int col = n0 + t * 16 + r16;
        const float bv = bias[col];
        #pragma unroll
        for (int r = 0; r < 8; ++r) {
            float v = accs[t][r] + bv;
            if (do_relu) v = fmaxf(v, 0.0f);
            C[(size_t)(mBase + r) * N + col] = v;
        }
    }
}

// ---------------------------------------------------------------------------
// Row inverse norms: inv_norm[i] = rsqrt(sum_d h[i,d]^2 + 1e-12). Wave per row.
// ---------------------------------------------------------------------------
__global__ void row_invnorm_kernel(const float* __restrict__ h,
                                   float* __restrict__ inv_norm, int n)
{
    const int lane = threadIdx.x & 31;
    const int row  = (blockIdx.x * blockDim.x + threadIdx.x) >> 5;
    if (row >= n) return;
    const float* hp = h + (size_t)row * HIDDEN;
    float s = 0.f;
    #pragma unroll
    for (int j = 0; j < HIDDEN / 32; ++j) {
        float v = hp[lane + j * 32];
        s += v * v;
    }
    #pragma unroll
    for (int off = 16; off > 0; off >>= 1)
        s += __shfl_xor(s, off, 32);
    if (lane == 0) inv_norm[row] = rsqrtf(s + 1e-12f);
}

// ---------------------------------------------------------------------------
// Per-edge: logit = cos(h_src, h_dst) in [-1,1]; e = exp(logit) (no max needed
// since bounded); accumulate z[dst]. Wave per edge. Edges [E, E+N) = self loops.
// ---------------------------------------------------------------------------
__global__ void edge_logit_kernel(const float* __restrict__ h,
                                  const float* __restrict__ inv_norm,
                                  const int* __restrict__ esrc,
                                  const int* __restrict__ edst,
                                  float* __restrict__ e_val,
                                  float* __restrict__ z, int etot)
{
    const int lane = threadIdx.x & 31;
    const int e    = (blockIdx.x * blockDim.x + threadIdx.x) >> 5;
    if (e >= etot) return;
    int s, d;
    if (e < N_EDGES) { s = esrc[e]; d = edst[e]; }
    else             { s = e - N_EDGES; d = s; }

    const float* hs = h + (size_t)s * HIDDEN;
    const float* hd = h + (size_t)d * HIDDEN;
    float acc = 0.f;
    #pragma unroll
    for (int j = 0; j < HIDDEN / 32; ++j)
        acc += hs[lane + j * 32] * hd[lane + j * 32];
    #pragma unroll
    for (int off = 16; off > 0; off >>= 1)
        acc += __shfl_xor(acc, off, 32);
    if (lane == 0) {
        float lg = acc * inv_norm[s] * inv_norm[d];   // cosine, |lg| <= ~1
        float ev = __expf(lg);
        e_val[e] = ev;
        atomicAdd(&z[d], ev);
    }
}

// ---------------------------------------------------------------------------
// Aggregate: out[dst] += (e_val/z[dst]) * h[src]. Wave per edge, 8 dims/lane.
// ---------------------------------------------------------------------------
__global__ void edge_agg_kernel(const float* __restrict__ h,
                                const int* __restrict__ esrc,
                                const int* __restrict__ edst,
                                const float* __restrict__ e_val,
                                const float* __restrict__ z,
                                float* __restrict__ out, int etot)
{
    const int lane = threadIdx.x & 31;
    const int e    = (blockIdx.x * blockDim.x + threadIdx.x) >> 5;
    if (e >= etot) return;
    int s, d;
    if (e < N_EDGES) { s = esrc[e]; d = edst[e]; }
    else             { s = e - N_EDGES; d = s; }

    float alpha = e_val[e] / z[d];                    // broadcast load (same line)
    const float* hs = h + (size_t)s * HIDDEN;
    float* od = out + (size_t)d * HIDDEN;
    #pragma unroll
    for (int j = 0; j < HIDDEN / 32; ++j)
        atomicAdd(&od[lane + j * 32], alpha * hs[lane + j * 32]);
}

__global__ void relu_kernel(float* __restrict__ h, int n)
{
    int i = blockIdx.x * blockDim.x + threadIdx.x;
    if (i < n) h[i] = fmaxf(h[i], 0.0f);
}

// ---------------------------------------------------------------------------
extern "C" void kernel_launch(void* const* d_in, const int* in_sizes, int n_in,
                              void* d_out, int out_size, void* d_ws, size_t ws_size,
                              hipStream_t stream)
{
    const float* x    = (const float*)d_in[0];   // [20000,128]
    const int*   esrc = (const int*)  d_in[1];   // [640000]
    const int*   edst = (const int*)  d_in[2];   // [640000]
    const float* W1   = (const float*)d_in[3];   // [128,256]
    const float* b1   = (const float*)d_in[4];   // [256]
    const float* W2   = (const float*)d_in[5];   // [256,64]
    const float* b2   = (const float*)d_in[6];   // [64]
    float*       out  = (float*)d_out;           // [20000,64]

    // Workspace layout (floats): hA | hB | inv_norm | z | e_val
    float* hA       = (float*)d_ws;
    float* hB       = hA + (size_t)N_NODES * HIDDEN;
    float* inv_norm = hB + (size_t)N_NODES * HIDDEN;
    float* zbuf     = inv_norm + N_NODES;
    float* e_val    = zbuf + N_NODES;

    const int ETOT = N_EDGES + N_NODES;          // graph edges + self loops

    // --- h = relu(x @ W1 + b1) ---------------------------------------------
    {
        int waves  = (N_NODES / 16) * (HIDDEN / 64);   // 5000
        int blocks = (waves * 32 + 255) / 256;
        gemm_wmma_f32_kernel<<<blocks, 256, 0, stream>>>(
            x, W1, b1, hA, N_NODES, IN_DIM, HIDDEN, 1);
    }

    // --- 4 AGNN layers ------------------------------------------------------
    float* hin  = hA;
    float* hout = hB;
    const int edgeBlocks = (ETOT * 32 + 255) / 256;
    const int rowBlocks  = (N_NODES * 32 + 255) / 256;
    const int eltBlocks  = (N_NODES * HIDDEN + 255) / 256;

    for (int layer = 0; layer < 4; ++layer) {
        hipMemsetAsync(zbuf, 0, (size_t)N_NODES * sizeof(float), stream);
        hipMemsetAsync(hout, 0, (size_t)N_NODES * HIDDEN * sizeof(float), stream);

        row_invnorm_kernel<<<rowBlocks, 256, 0, stream>>>(hin, inv_norm, N_NODES);
        edge_logit_kernel<<<edgeBlocks, 256, 0, stream>>>(
            hin, inv_norm, esrc, edst, e_val, zbuf, ETOT);
        edge_agg_kernel<<<edgeBlocks, 256, 0, stream>>>(
            hin, esrc, edst, e_val, zbuf, hout, ETOT);
        relu_kernel<<<eltBlocks, 256, 0, stream>>>(hout, N_NODES * HIDDEN);

        float* t = hin; hin = hout; hout = t;    // after 4 swaps, final h in hA
    }

    // --- out = h @ W2 + b2 --------------------------------------------------
    {
        int waves  = (N_NODES / 16) * (OUT_DIM / 64); // 1250
        int blocks = (waves * 32 + 255) / 256;
        gemm_wmma_f32_kernel<<<blocks, 256, 0, stream>>>(
            hin, W2, b2, out, N_NODES, HIDDEN, OUT_DIM, 0);
    }
}